// mytransformer_9655086481757
// MI455X (gfx1250) — compile-verified
//
#include <hip/hip_runtime.h>
#include <hip/hip_bf16.h>

// ---------------------------------------------------------------------------
// CDNA5 (gfx1250) f32 WMMA transformer-encoder scan.
//   B=128, T=24, d=128, dff=2048, L=2, nhead=4, dh=32, din=512, nc=64
// All matmuls use V_WMMA_F32_16X16X4_F32 with 2x2 register blocking (2x1, a 32x16 tile, in projln/ff)
// (32x32 macro-tile per wave: 1 load per wmma, 4 independent acc chains).
// ---------------------------------------------------------------------------

typedef float f32x2 __attribute__((ext_vector_type(2)));
typedef float f32x8 __attribute__((ext_vector_type(8)));

#define B_    128
#define T_    24
#define D_    128
#define BT_   3072          // B_*T_
#define DFF_  2048
#define NH_   4
#define DH_   32
#define NC_   64

__device__ __forceinline__ f32x8 wmma4(f32x2 a, f32x2 b, f32x8 c) {
  // D = A(16x4,f32) * B(4x16,f32) + C(16x16,f32)
  return __builtin_amdgcn_wmma_f32_16x16x4_f32(false, a, false, b, (short)0, c,
                                               false, false);
}

__device__ __forceinline__ f32x2 ldv2(const float* p) {
  return *(const f32x2*)p;   // 8B-aligned by construction (even strides/kk)
}

// 2x2-blocked K-loop: wave computes a 32x32 tile of A @ W^T.
// a0/a1 = lane's A rows (m and m+16), w0/w1 = lane's W rows (n and n+16).
__device__ __forceinline__ void gemm_2x2(const float* a0, const float* a1,
                                         const float* w0, const float* w1,
                                         int K, int half,
                                         f32x8& c00, f32x8& c01,
                                         f32x8& c10, f32x8& c11) {
#pragma unroll 8
  for (int k = 0; k < K; k += 4) {
    int kk = k + (half << 1);
    f32x2 fa0 = ldv2(a0 + kk), fa1 = ldv2(a1 + kk);
    f32x2 fb0 = ldv2(w0 + kk), fb1 = ldv2(w1 + kk);
    c00 = wmma4(fa0, fb0, c00);
    c01 = wmma4(fa0, fb1, c01);
    c10 = wmma4(fa1, fb0, c10);
    c11 = wmma4(fa1, fb1, c11);
  }
}

// 2x1-blocked K-loop: 32x16 tile (two row blocks, one col block).
__device__ __forceinline__ void gemm_2x1(const float* a0, const float* a1,
                                         const float* w0, int K, int half,
                                         f32x8& c0, f32x8& c1) {
#pragma unroll 8
  for (int k = 0; k < K; k += 4) {
    int kk = k + (half << 1);
    f32x2 fa0 = ldv2(a0 + kk), fa1 = ldv2(a1 + kk);
    f32x2 fb0 = ldv2(w0 + kk);
    c0 = wmma4(fa0, fb0, c0);
    c1 = wmma4(fa1, fb0, c1);
  }
}

// LayerNorm epilogue over a 32x128 tile in LDS (stride 130 floats).
// Block = 256 threads.  mrs = 64 floats of LDS.
__device__ __forceinline__ void ln_epilogue32(float (*tile)[130], float* mrs,
                                              const float* gam, const float* bet,
                                              float* out, int tm, int tid) {
  if (tid < 32) {
    float s = 0.f;
#pragma unroll 8
    for (int c = 0; c < D_; ++c) s += tile[tid][c];
    float m = s * (1.f / 128.f);
    float v = 0.f;
#pragma unroll 8
    for (int c = 0; c < D_; ++c) { float d = tile[tid][c] - m; v += d * d; }
    v *= (1.f / 128.f);
    mrs[tid * 2 + 0] = m;
    mrs[tid * 2 + 1] = rsqrtf(v + 1e-5f);
  }
  __syncthreads();
#pragma unroll
  for (int e = 0; e < 16; ++e) {
    int idx = e * 256 + tid;           // coalesced
    int rr = idx >> 7, cc = idx & 127;
    float val = (tile[rr][cc] - mrs[rr * 2]) * mrs[rr * 2 + 1] * gam[cc] + bet[cc];
    out[(size_t)(tm + rr) * D_ + cc] = val;
  }
}

// ---------------------------------------------------------------------------
// Kernel 1: h0 = relu(x @ Wfc^T + bfc) + posenc.   M=3072, N=128, K=512.
// 32x32 macro-tiles: 96*4 = 384 wave-tiles -> 48 blocks x 256 threads.
// ---------------------------------------------------------------------------
__global__ void fc_pe_kernel(const float* __restrict__ x,
                             const float* __restrict__ Wfc,
                             const float* __restrict__ bfc,
                             float* __restrict__ h0) {
  int wave = threadIdx.x >> 5, lane = threadIdx.x & 31;
  int half = lane >> 4, low = lane & 15;
  int tile = blockIdx.x * 8 + wave;     // ntn32 = 4
  int tm = (tile >> 2) << 5;
  int tn = (tile & 3) << 5;
  const float* a0 = x + (size_t)(tm + low) * 512;
  const float* a1 = x + (size_t)(tm + 16 + low) * 512;
  const float* w0 = Wfc + (size_t)(tn + low) * 512;
  const float* w1 = Wfc + (size_t)(tn + 16 + low) * 512;
  f32x8 c00 = {}, c01 = {}, c10 = {}, c11 = {};
  gemm_2x2(a0, a1, w0, w1, 512, half, c00, c01, c10, c11);
  const float lg = logf(10000.f) / 128.f;
#pragma unroll
  for (int cb = 0; cb < 2; ++cb) {
    int n = tn + cb * 16 + low;
    float bv = bfc[n];
    float freq = expf(-(float)(n & ~1) * lg);
#pragma unroll
    for (int rb = 0; rb < 2; ++rb) {
      f32x8 acc = cb == 0 ? (rb == 0 ? c00 : c10) : (rb == 0 ? c01 : c11);
#pragma unroll
      for (int r = 0; r < 8; ++r) {
        int row = tm + rb * 16 + r + (half << 3);
        float arg = (float)(row % T_) * freq;
        float pe = (n & 1) ? cosf(arg) : sinf(arg);
        h0[(size_t)row * D_ + n] = fmaxf(acc[r] + bv, 0.f) + pe;
      }
    }
  }
}

// ---------------------------------------------------------------------------
// Kernel 2: generic GEMM  C = A @ W^T + bias  (optional relu).
// A rows taken modulo aMod.  32x32 macro-tile per wave.
// Grid: tiles32/8 blocks x 256 threads; tiles32 = (M/32)*(N/32).
// ---------------------------------------------------------------------------
__global__ void gemm_bias_kernel(const float* __restrict__ A, int aMod,
                                 const float* __restrict__ W,
                                 const float* __restrict__ bias,
                                 float* __restrict__ C,
                                 int N, int K, int doRelu) {
  int wave = threadIdx.x >> 5, lane = threadIdx.x & 31;
  int half = lane >> 4, low = lane & 15;
  int tile = blockIdx.x * 8 + wave;
  int ntn = N >> 5;
  int tm = (tile / ntn) << 5;
  int tn = (tile % ntn) << 5;
  int am = tm % aMod;                 // aMod is a multiple of 32
  const float* a0 = A + (size_t)(am + low) * K;
  const float* a1 = A + (size_t)(am + 16 + low) * K;
  const float* w0 = W + (size_t)(tn + low) * K;
  const float* w1 = W + (size_t)(tn + 16 + low) * K;
  f32x8 c00 = {}, c01 = {}, c10 = {}, c11 = {};
  gemm_2x2(a0, a1, w0, w1, K, half, c00, c01, c10, c11);
#pragma unroll
  for (int cb = 0; cb < 2; ++cb) {
    float bv = bias[tn + cb * 16 + low];
#pragma unroll
    for (int rb = 0; rb < 2; ++rb) {
      f32x8 acc = cb == 0 ? (rb == 0 ? c00 : c10) : (rb == 0 ? c01 : c11);
#pragma unroll
      for (int r = 0; r < 8; ++r) {
        float v = acc[r] + bv;
        if (doRelu) v = fmaxf(v, 0.f);
        C[(size_t)(tm + rb * 16 + r + (half << 3)) * N + tn + cb * 16 + low] = v;
      }
    }
  }
}

// ---------------------------------------------------------------------------
// Kernel 3: attention per (prefix g, batch b, head h).  keys <= istart+g.
// qkv rows: token (g,b,t), 384 cols = [q | k | v], head h at cols h*32.
// Grid: Gc*512 blocks x 32 threads (thread = query position).
// ---------------------------------------------------------------------------
__global__ void attn_kernel(const float* __restrict__ qkv,
                            float* __restrict__ ctx, int istart) {
  int blk = blockIdx.x;
  int h = blk & 3;
  int b = (blk >> 2) & 127;
  int g = blk >> 9;
  int i = istart + g;
  int t = threadIdx.x;
  if (t >= T_) return;
  const float* base = qkv + (size_t)(g * B_ + b) * T_ * 384;
  float q[DH_];
#pragma unroll
  for (int n = 0; n < DH_; ++n) q[n] = base[t * 384 + h * DH_ + n];
  float sc[T_];
  float mx = -1e30f;
  for (int k = 0; k <= i; ++k) {
    const float* kr = base + k * 384 + 128 + h * DH_;
    float s = 0.f;
#pragma unroll
    for (int n = 0; n < DH_; ++n) s += q[n] * kr[n];
    s *= 0.17677669529663688f;   // 1/sqrt(32)
    sc[k] = s;
    mx = fmaxf(mx, s);
  }
  float sum = 0.f;
  for (int k = 0; k <= i; ++k) { float e = expf(sc[k] - mx); sc[k] = e; sum += e; }
  float inv = 1.f / sum;
  float acc[DH_];
#pragma unroll
  for (int n = 0; n < DH_; ++n) acc[n] = 0.f;
  for (int k = 0; k <= i; ++k) {
    const float* vr = base + k * 384 + 256 + h * DH_;
    float p = sc[k] * inv;
#pragma unroll
    for (int n = 0; n < DH_; ++n) acc[n] += p * vr[n];
  }
  float* crow = ctx + ((size_t)(g * B_ + b) * T_ + t) * D_ + h * DH_;
#pragma unroll
  for (int n = 0; n < DH_; ++n) crow[n] = acc[n];
}

// ---------------------------------------------------------------------------
// Kernel 4: out = LN( resid + ctx @ Wo^T + bo ).  Per 32-row tile.
// Block = 256 threads; wave w owns output cols [w*16, w*16+16) for 32 rows.
// ---------------------------------------------------------------------------
__global__ void projln_kernel(const float* __restrict__ ctx,
                              const float* __restrict__ Wo,
                              const float* __restrict__ bo,
                              const float* __restrict__ resid, int rMod,
                              const float* __restrict__ gam,
                              const float* __restrict__ bet,
                              float* __restrict__ out) {
  __shared__ float tile[32][130];
  __shared__ float mrs[64];
  int tid = threadIdx.x;
  int wave = tid >> 5, lane = tid & 31;
  int half = lane >> 4, low = lane & 15;
  int tm = blockIdx.x << 5;
  int n0 = wave << 4;
  const float* a0 = ctx + (size_t)(tm + low) * D_;
  const float* a1 = ctx + (size_t)(tm + 16 + low) * D_;
  const float* w0 = Wo + (size_t)(n0 + low) * D_;
  f32x8 c0 = {}, c1 = {};
  gemm_2x1(a0, a1, w0, D_, half, c0, c1);
  float bv = bo[n0 + low];
  int rbase = tm % rMod;                // rMod multiple of 32
#pragma unroll
  for (int r = 0; r < 8; ++r) {
    int rr = r + (half << 3);
    tile[rr][n0 + low]      = c0[r] + bv + resid[(size_t)(rbase + rr) * D_ + n0 + low];
    tile[16 + rr][n0 + low] = c1[r] + bv + resid[(size_t)(rbase + 16 + rr) * D_ + n0 + low];
  }
  __syncthreads();
  ln_epilogue32(tile, mrs, gam, bet, out, tm, tid);
}

// ---------------------------------------------------------------------------
// Kernel 5: fused FFN:  out = LN( y + relu(y @ W1^T + b1) @ W2^T + b2 ).
// Per 32-row tile.  Each round, 8 waves cooperatively stage 128 hidden dims
// (relu-applied) through LDS; then every wave accumulates its own 32x16
// output-column block over the staged K=128 (LDS A-operands).
// Deterministic: no atomics, wave w owns output cols [w*16, w*16+16).
// ---------------------------------------------------------------------------
__global__ void ff_kernel(const float* __restrict__ y,
                          const float* __restrict__ W1,
                          const float* __restrict__ b1,
                          const float* __restrict__ W2,
                          const float* __restrict__ b2,
                          const float* __restrict__ gam,
                          const float* __restrict__ bet,
                          float* __restrict__ out) {
  __shared__ float hst[32][130];
  __shared__ float tile[32][130];
  __shared__ float mrs[64];
  int tid = threadIdx.x;
  int wave = tid >> 5, lane = tid & 31;
  int half = lane >> 4, low = lane & 15;
  int tm = blockIdx.x << 5;
  int n0 = wave << 4;
  const float* a0 = y + (size_t)(tm + low) * D_;
  const float* a1 = y + (size_t)(tm + 16 + low) * D_;
  f32x8 f0 = {}, f1 = {};
  for (int rnd = 0; rnd < DFF_ / 128; ++rnd) {
    int hid0 = rnd * 128 + (wave << 4);
    // hidden chunk: h = relu(y @ W1^T + b1), 32 rows x 16 hidden
    const float* w1row = W1 + (size_t)(hid0 + low) * D_;
    f32x8 h0a = {}, h1a = {};
    gemm_2x1(a0, a1, w1row, D_, half, h0a, h1a);
    // prefetch next round's weight rows into L2/WGP$ (global_prefetch_b8)
    if (rnd + 1 < DFF_ / 128) {
      __builtin_prefetch(W1 + (size_t)(hid0 + 128 + low) * D_, 0, 1);
      __builtin_prefetch(W2 + (size_t)(n0 + low) * DFF_ + (rnd + 1) * 128, 0, 1);
    }
    float bv1 = b1[hid0 + low];
    __syncthreads();   // previous round's readers done before overwrite
#pragma unroll
    for (int r = 0; r < 8; ++r) {
      hst[r + (half << 3)][(wave << 4) + low]      = fmaxf(h0a[r] + bv1, 0.f);
      hst[16 + r + (half << 3)][(wave << 4) + low] = fmaxf(h1a[r] + bv1, 0.f);
    }
    __syncthreads();
    // f += h_staged @ W2^T over this 128-wide hidden block (A from LDS)
    const float* w2row = W2 + (size_t)(n0 + low) * DFF_ + rnd * 128;
    const float* h0row = &hst[low][0];
    const float* h1row = &hst[16 + low][0];
    gemm_2x1(h0row, h1row, w2row, 128, half, f0, f1);
  }
  float bv2 = b2[n0 + low];
#pragma unroll
  for (int r = 0; r < 8; ++r) {
    int rr = r + (half << 3);
    tile[rr][n0 + low]      = f0[r] + bv2 + y[(size_t)(tm + rr) * D_ + n0 + low];
    tile[16 + rr][n0 + low] = f1[r] + bv2 + y[(size_t)(tm + 16 + rr) * D_ + n0 + low];
  }
  __syncthreads();
  ln_epilogue32(tile, mrs, gam, bet, out, tm, tid);
}

// ---------------------------------------------------------------------------
// Kernel 6: gather row i of each prefix run into enc_out[b*T + i].
// threads = Gc * B_ * D_.
// ---------------------------------------------------------------------------
__global__ void extract_kernel(const float* __restrict__ bufB,
                               float* __restrict__ enc_out, int istart) {
  int idx = blockIdx.x * 256 + threadIdx.x;
  int n = idx & 127;
  int rb = idx >> 7;          // g*128 + b
  int b = rb & 127;
  int g = rb >> 7;
  int i = istart + g;
  enc_out[((size_t)b * T_ + i) * D_ + n] =
      bufB[(((size_t)(g * B_ + b)) * T_ + i) * D_ + n];
}

// ---------------------------------------------------------------------------
// Kernel 7: logits = chid @ Wc2^T + bc2 ; out = log_softmax(logits).
// One thread per token row (3072).
// ---------------------------------------------------------------------------
__global__ void cls2_kernel(const float* __restrict__ chid,
                            const float* __restrict__ Wc2,
                            const float* __restrict__ bc2,
                            float* __restrict__ out) {
  int row = blockIdx.x * 256 + threadIdx.x;
  if (row >= BT_) return;
  const float* h = chid + (size_t)row * NC_;
  float l0 = bc2[0], l1 = bc2[1];
#pragma unroll 8
  for (int k = 0; k < NC_; ++k) {
    l0 += h[k] * Wc2[k];
    l1 += h[k] * Wc2[NC_ + k];
  }
  float m = fmaxf(l0, l1);
  float lse = m + logf(expf(l0 - m) + expf(l1 - m));
  out[(size_t)row * 2 + 0] = l0 - lse;
  out[(size_t)row * 2 + 1] = l1 - lse;
}

// ---------------------------------------------------------------------------
extern "C" void kernel_launch(void* const* d_in, const int* in_sizes, int n_in,
                              void* d_out, int out_size, void* d_ws, size_t ws_size,
                              hipStream_t stream) {
  (void)in_sizes; (void)n_in; (void)out_size;
  const float* x    = (const float*)d_in[0];
  const float* Wfc  = (const float*)d_in[1];
  const float* bfc  = (const float*)d_in[2];
  const float* qkvw = (const float*)d_in[3];
  const float* qkvb = (const float*)d_in[4];
  const float* outw = (const float*)d_in[5];
  const float* outb = (const float*)d_in[6];
  const float* ln1g = (const float*)d_in[7];
  const float* ln1b = (const float*)d_in[8];
  const float* ln2g = (const float*)d_in[9];
  const float* ln2b = (const float*)d_in[10];
  const float* ff1w = (const float*)d_in[11];
  const float* ff1b = (const float*)d_in[12];
  const float* ff2w = (const float*)d_in[13];
  const float* ff2b = (const float*)d_in[14];
  const float* Wc1  = (const float*)d_in[15];
  const float* bc1  = (const float*)d_in[16];
  const float* Wc2  = (const float*)d_in[17];
  const float* bc2  = (const float*)d_in[18];

  float* ws = (float*)d_ws;
  // Fixed-size regions
  float* h0      = ws;                     // 3072*128
  float* enc_out = h0 + (size_t)BT_ * D_;  // 3072*128
  float* chid    = enc_out + (size_t)BT_ * D_;  // 3072*64
  float* dyn     = chid + (size_t)BT_ * NC_;

  // Pick prefix-batch factor G (deterministic in ws_size): largest G<=24 s.t.
  // workspace fits.  Per-G floats: G*3072*(384 + 3*128) = G*2359296.
  const size_t fixedF = (size_t)BT_ * D_ * 2 + (size_t)BT_ * NC_;
  int G = 24;
  while (G > 1) {
    size_t needBytes = (fixedF + (size_t)G * BT_ * 768) * sizeof(float);
    if (needBytes <= ws_size) break;
    --G;
  }

  // h0 = relu(x @ Wfc^T + bfc) + posenc
  fc_pe_kernel<<<48, 256, 0, stream>>>(x, Wfc, bfc, h0);

  for (int s = 0; s < T_; s += G) {
    int Gc = (T_ - s < G) ? (T_ - s) : G;
    size_t MG = (size_t)Gc * BT_;
    float* qkv  = dyn;
    float* ctxb = qkv  + MG * 384;
    float* bufA = ctxb + MG * D_;
    float* bufB = bufA + MG * D_;
    for (int l = 0; l < 2; ++l) {
      const float* yin = (l == 0) ? h0 : bufB;
      int aMod = (l == 0) ? BT_ : (int)MG;
      // qkv = yin @ qkv_w^T + qkv_b   (M=Gc*3072, N=384, K=128)
      // tiles32 = (M/32)*(384/32) = Gc*96*12 -> Gc*144 blocks
      gemm_bias_kernel<<<Gc * 144, 256, 0, stream>>>(
          yin, aMod, qkvw + (size_t)l * 384 * D_, qkvb + (size_t)l * 384,
          qkv, 384, D_, 0);
      // masked multi-head attention -> ctx
      attn_kernel<<<Gc * 512, 32, 0, stream>>>(qkv, ctxb, s);
      // bufA = LN1( yin + ctx @ out_w^T + out_b )    (32-row tiles)
      projln_kernel<<<Gc * 96, 256, 0, stream>>>(
          ctxb, outw + (size_t)l * D_ * D_, outb + (size_t)l * D_,
          yin, aMod, ln1g + (size_t)l * D_, ln1b + (size_t)l * D_, bufA);
      // bufB = LN2( bufA + relu(bufA @ ff1^T + b1) @ ff2^T + b2 )
      ff_kernel<<<Gc * 96, 256, 0, stream>>>(
          bufA, ff1w + (size_t)l * DFF_ * D_, ff1b + (size_t)l * DFF_,
          ff2w + (size_t)l * D_ * DFF_, ff2b + (size_t)l * D_,
          ln2g + (size_t)l * D_, ln2b + (size_t)l * D_, bufB);
    }
    // collect row i of each prefix run
    extract_kernel<<<Gc * 64, 256, 0, stream>>>(bufB, enc_out, s);
  }

  // classifier: chid = relu(enc_out @ Wc1^T + bc1)  (M=3072, N=64, K=128)
  // tiles32 = 96*2 = 192 -> 24 blocks
  gemm_bias_kernel<<<24, 256, 0, stream>>>(enc_out, BT_, Wc1, bc1, chid, NC_, D_, 1);
  // logits + log_softmax -> d_out (3072 x 2, f32)
  cls2_kernel<<<12, 256, 0, stream>>>(chid, Wc2, bc2, (float*)d_out);
}